// BertEmbedding_41772851921356
// MI455X (gfx1250) — compile-verified
//
#include <hip/hip_runtime.h>

// BERT embedding (gather x3 + add) fused with LayerNorm(H=768), fp32.
// B=32, S=512, H=768 -> 16384 rows. Memory-bound: ~200MB traffic, ~8.6us
// floor at 23.3 TB/s HBM. One wave32 per row; 6x float4 per lane per table.

typedef float v4f __attribute__((ext_vector_type(4)));

#define HIDDEN   768
#define SEQ_LEN  512
#define CHUNKS   6          // 768 cols / (32 lanes * 4 floats)
#define WAVES_PER_BLOCK 8
#define EPS      1e-5f

__global__ __launch_bounds__(256) void bert_emb_ln_kernel(
    const int*   __restrict__ input_ids,      // [B*S]
    const int*   __restrict__ token_type_ids, // [B*S]
    const float* __restrict__ tok_w,          // [VOCAB, H]
    const float* __restrict__ pos_w,          // [MAX_POS, H]
    const float* __restrict__ type_w,         // [TYPE_VOCAB, H]
    const float* __restrict__ gamma,          // [H]
    const float* __restrict__ beta,           // [H]
    float*       __restrict__ out,            // [B*S, H]
    int n_rows)
{
    const int wave = threadIdx.x >> 5;             // 8 waves per block
    const int lane = threadIdx.x & 31;             // wave32 (gfx1250)
    const int row  = blockIdx.x * WAVES_PER_BLOCK + wave;
    if (row >= n_rows) return;

    const int s_pos = row & (SEQ_LEN - 1);         // position id (S == 512)
    const int tok   = input_ids[row];
    const int typ   = token_type_ids[row];

    const v4f* __restrict__ tw = (const v4f*)(tok_w  + (size_t)tok   * HIDDEN);
    const v4f* __restrict__ pw = (const v4f*)(pos_w  + (size_t)s_pos * HIDDEN);
    const v4f* __restrict__ yw = (const v4f*)(type_w + (size_t)typ   * HIDDEN);

    // Warm the randomly-gathered vocab row (global_prefetch_b8) before the
    // dependent b128 loads issue; pos_w/type_w are L2-resident (1.5MB/6KB).
    __builtin_prefetch((const void*)(tw + lane), 0, 1);

    // One pass: emb = tok + pos + type, accumulate sum and sum-of-squares.
    v4f acc[CHUNKS];
    float s1 = 0.0f, s2 = 0.0f;
    #pragma unroll
    for (int c = 0; c < CHUNKS; ++c) {
        const int idx = c * 32 + lane;             // coalesced 512B segments
        v4f v = tw[idx] + pw[idx] + yw[idx];
        acc[c] = v;
        s1 += v.x + v.y + v.z + v.w;
        s2 += v.x * v.x + v.y * v.y + v.z * v.z + v.w * v.w;
    }

    // Wave32 butterfly reduction (DS-permute path on gfx1250).
    #pragma unroll
    for (int off = 16; off > 0; off >>= 1) {
        s1 += __shfl_xor(s1, off, 32);
        s2 += __shfl_xor(s2, off, 32);
    }

    const float inv_h = 1.0f / (float)HIDDEN;
    const float mean  = s1 * inv_h;
    const float var   = fmaxf(s2 * inv_h - mean * mean, 0.0f);
    const float rstd  = rsqrtf(var + EPS);

    const v4f* __restrict__ g4 = (const v4f*)gamma;
    const v4f* __restrict__ b4 = (const v4f*)beta;
    v4f* __restrict__ o4 = (v4f*)(out + (size_t)row * HIDDEN);

    // Normalize + affine; non-temporal b128 stores (output never re-read,
    // keep the 192MB L2 for the vocab-gather working set instead).
    #pragma unroll
    for (int c = 0; c < CHUNKS; ++c) {
        const int idx = c * 32 + lane;
        v4f g = g4[idx];
        v4f b = b4[idx];
        v4f v = acc[c];
        v4f r;
        r.x = (v.x - mean) * rstd * g.x + b.x;
        r.y = (v.y - mean) * rstd * g.y + b.y;
        r.z = (v.z - mean) * rstd * g.z + b.z;
        r.w = (v.w - mean) * rstd * g.w + b.w;
        __builtin_nontemporal_store(r, o4 + idx);
    }
}

extern "C" void kernel_launch(void* const* d_in, const int* in_sizes, int n_in,
                              void* d_out, int out_size, void* d_ws, size_t ws_size,
                              hipStream_t stream) {
    // setup_inputs() order:
    //   0: input_ids [B,S] int32
    //   1: token_type_ids [B,S] int32
    //   2: tok_w  [VOCAB, H]   f32
    //   3: pos_w  [MAX_POS, H] f32
    //   4: type_w [TYPE_VOCAB, H] f32
    //   5: gamma [H] f32
    //   6: beta  [H] f32
    const int*   input_ids      = (const int*)  d_in[0];
    const int*   token_type_ids = (const int*)  d_in[1];
    const float* tok_w          = (const float*)d_in[2];
    const float* pos_w          = (const float*)d_in[3];
    const float* type_w         = (const float*)d_in[4];
    const float* gamma          = (const float*)d_in[5];
    const float* beta           = (const float*)d_in[6];
    float*       out            = (float*)d_out;

    const int n_rows = in_sizes[0];                     // B*S = 16384
    const int blocks = (n_rows + WAVES_PER_BLOCK - 1) / WAVES_PER_BLOCK;

    bert_emb_ln_kernel<<<blocks, 256, 0, stream>>>(
        input_ids, token_type_ids, tok_w, pos_w, type_w, gamma, beta,
        out, n_rows);
}